// GridSampler_67997922230580
// MI455X (gfx1250) — compile-verified
//
#include <hip/hip_runtime.h>
#include <stdint.h>

#ifndef __has_builtin
#define __has_builtin(x) 0
#endif

// ---------------- reference constants ----------------
#define G0 1.0f
#define G1 0.4f
#define PLOX -50.0f
#define PLOY -50.0f
#define PLOZ -4.0f
#define PHIX 50.0f
#define PHIY 50.0f
#define PHIZ 4.0f

// Key-space bound: num_b<=9 (b in [0,8)), 251 x 251 x 21 spatial cells.
// 9*251*251*21 = 11,907,189 cells -> pad to whole scan chunks.
#define CHUNK 8192
#define NCHUNKS 1454
#define M_PAD (NCHUNKS * CHUNK)   // 11,911,168 cells, 47.6 MB (fits in 192MB L2)

#define AS_GLOBAL __attribute__((address_space(1)))
#define AS_LDS    __attribute__((address_space(3)))

typedef int vec4i __attribute__((vector_size(16)));

// ---------------- CDNA5 async global->LDS copy ----------------
__device__ __forceinline__ void async_copy_b128(const void* gsrc, void* ldst) {
#if __has_builtin(__builtin_amdgcn_global_load_async_to_lds_b128)
  __builtin_amdgcn_global_load_async_to_lds_b128(
      (AS_GLOBAL vec4i*)(uintptr_t)gsrc,
      (AS_LDS vec4i*)(uint32_t)(uintptr_t)ldst, 0, 0);
#else
  unsigned lo = (unsigned)(uintptr_t)ldst;               // LDS byte offset (addr[31:0])
  unsigned long long ga = (unsigned long long)(uintptr_t)gsrc;
  asm volatile("global_load_async_to_lds_b128 %0, %1, off"
               :: "v"(lo), "v"(ga) : "memory");
#endif
}
#define WAIT_ASYNCCNT_0() asm volatile("s_wait_asynccnt 0" ::: "memory")
#define WAIT_ASYNCCNT_1() asm volatile("s_wait_asynccnt 1" ::: "memory")

// ---------------- ordered-int float min/max encoding ----------------
__device__ __forceinline__ unsigned f2ord(float f) {
  unsigned u = __float_as_uint(f);
  return (u & 0x80000000u) ? ~u : (u | 0x80000000u);
}
__device__ __forceinline__ float ord2f(unsigned u) {
  u = (u & 0x80000000u) ? (u ^ 0x80000000u) : ~u;
  return __uint_as_float(u);
}

// ---------------- block-wide exclusive scan (256 thr, wave32) ----------------
__device__ __forceinline__ int block_excl_scan_256(int v, int* lds, int* total) {
  const int lane = threadIdx.x & 31;
  const int wid  = threadIdx.x >> 5;
  int inc = v;
#pragma unroll
  for (int d = 1; d < 32; d <<= 1) {
    int y = __shfl_up(inc, d, 32);
    if (lane >= d) inc += y;
  }
  if (lane == 31) lds[wid] = inc;
  __syncthreads();
  if (threadIdx.x < 32) {
    int w = (threadIdx.x < 8) ? lds[threadIdx.x] : 0;
#pragma unroll
    for (int d = 1; d < 8; d <<= 1) {
      int y = __shfl_up(w, d, 32);
      if (lane >= d) w += y;
    }
    if (threadIdx.x < 8) lds[8 + threadIdx.x] = w;
  }
  __syncthreads();
  int waveExcl = wid ? lds[8 + wid - 1] : 0;
  *total = lds[15];
  return waveExcl + (inc - v);
}

// ================= K0: zero flags/counts/sums, init minmax =================
__global__ void k_init(int* __restrict__ flags, float* __restrict__ counts,
                       float* __restrict__ sums4, unsigned* __restrict__ mm, int n) {
  size_t i = (size_t)blockIdx.x * blockDim.x + threadIdx.x;
  size_t stride = (size_t)gridDim.x * blockDim.x;
  for (size_t j = i; j < (size_t)M_PAD; j += stride) flags[j] = 0;
  for (size_t j = i; j < (size_t)n; j += stride) counts[j] = 0.0f;
  for (size_t j = i; j < (size_t)n * 4; j += stride) sums4[j] = 0.0f;
  if (i == 0) { mm[0] = 0xFFFFFFFFu; mm[1] = 0u; }
}

// ================= K1: min/max of batch column =================
__global__ void k_minmax(const float* __restrict__ pts, unsigned* __restrict__ mm, int n) {
  size_t stride = (size_t)gridDim.x * blockDim.x;
  unsigned lmin = 0xFFFFFFFFu, lmax = 0u;
  for (size_t i = (size_t)blockIdx.x * blockDim.x + threadIdx.x; i < (size_t)n; i += stride) {
    unsigned e = f2ord(pts[i * 4]);
    lmin = min(lmin, e);
    lmax = max(lmax, e);
  }
#pragma unroll
  for (int d = 16; d >= 1; d >>= 1) {
    lmin = min(lmin, (unsigned)__shfl_xor((int)lmin, d, 32));
    lmax = max(lmax, (unsigned)__shfl_xor((int)lmax, d, 32));
  }
  if ((threadIdx.x & 31) == 0) {
    atomicMin(&mm[0], lmin);
    atomicMax(&mm[1], lmax);
  }
}

// ====== K2: cluster keys + presence flags (double-buffered async->LDS) ======
__global__ void k_cluster(const float4* __restrict__ pts, const unsigned* __restrict__ mm,
                          int* __restrict__ cluster, int* __restrict__ flags, int n) {
  __shared__ float4 buf[2][256];
  float bmin = ord2f(mm[0]), bmax = ord2f(mm[1]);
  float s0 = bmin - 0.5f;
  float e0 = bmax + 0.5f;
  int n0 = (int)floorf((e0 - s0) / G0) + 1;
  const int n1 = (int)floorf((PHIX - PLOX) / G1) + 1;  // 251
  const int n2 = (int)floorf((PHIY - PLOY) / G1) + 1;  // 251
  int st1 = n0, st2 = n0 * n1, st3 = n0 * n1 * n2;

  const int tid = threadIdx.x;
  const long stride = (long)gridDim.x * blockDim.x;
  long i = (long)blockIdx.x * blockDim.x + tid;
  if (i < n) async_copy_b128(&pts[i], &buf[0][tid]);
  int pb = 0;
  for (; i < n; i += stride) {
    long nx = i + stride;
    if (nx < n) {
      async_copy_b128(&pts[nx], &buf[pb ^ 1][tid]);  // prefetch next tile
      WAIT_ASYNCCNT_1();                             // current tile done
    } else {
      WAIT_ASYNCCNT_0();
    }
    float4 p = buf[pb][tid];
    int c0 = (int)floorf((p.x - s0) / G0);
    int c1 = (int)floorf((p.y - PLOX) / G1);
    int c2 = (int)floorf((p.z - PLOY) / G1);
    int c3 = (int)floorf((p.w - PLOZ) / G1);
    int cl = c0 + c1 * st1 + c2 * st2 + c3 * st3;
    cl = min(max(cl, 0), M_PAD - 1);
    cluster[i] = cl;
    flags[cl] = 1;                                   // idempotent presence mark
    pb ^= 1;
  }
}

// ================= K3: per-chunk flag totals =================
__global__ void k_chunksum(const int* __restrict__ flags, int* __restrict__ csum) {
  __shared__ int red[8];
  int base = blockIdx.x * CHUNK;
  int s = 0;
  for (int j = threadIdx.x; j < CHUNK; j += 256) s += flags[base + j];
#pragma unroll
  for (int d = 16; d >= 1; d >>= 1) s += __shfl_xor(s, d, 32);
  if ((threadIdx.x & 31) == 0) red[threadIdx.x >> 5] = s;
  __syncthreads();
  if (threadIdx.x == 0) {
    int t = 0;
#pragma unroll
    for (int w = 0; w < 8; ++w) t += red[w];
    csum[blockIdx.x] = t;
  }
}

// ================= K4: exclusive scan over chunk totals (1 block) =================
__global__ void k_scanchunks(const int* __restrict__ csum, int* __restrict__ coff) {
  __shared__ int lds[16];
  int carry = 0;
  for (int base = 0; base < NCHUNKS; base += 256) {
    int idx = base + threadIdx.x;
    int v = (idx < NCHUNKS) ? csum[idx] : 0;
    int total;
    int ex = block_excl_scan_256(v, lds, &total);
    if (idx < NCHUNKS) coff[idx] = carry + ex;
    __syncthreads();
    carry += total;
  }
}

// ====== K5: in-chunk inclusive ranks; flags[] becomes rank[] in place ======
__global__ void k_rank(int* __restrict__ flags, const int* __restrict__ coff) {
  __shared__ int tile[CHUNK];
  __shared__ int lds[16];
  const int base = blockIdx.x * CHUNK;
  for (int j = threadIdx.x; j < CHUNK; j += 256) tile[j] = flags[base + j];
  __syncthreads();
  const int SEQ = CHUNK / 256;  // 32
  const int t0 = threadIdx.x * SEQ;
  int s = 0;
#pragma unroll
  for (int j = 0; j < SEQ; ++j) s += tile[t0 + j];
  int total;
  int ex = block_excl_scan_256(s, lds, &total);
  int run = coff[blockIdx.x] + ex;
#pragma unroll
  for (int j = 0; j < SEQ; ++j) { run += tile[t0 + j]; tile[t0 + j] = run; }
  __syncthreads();
  for (int j = threadIdx.x; j < CHUNK; j += 256) flags[base + j] = tile[j];
}

// ================= K6: per-point inverse + scatter-mean accumulation =================
__global__ void k_scatter(const float4* __restrict__ pts, const int* __restrict__ cluster,
                          const int* __restrict__ rank, float* __restrict__ sums,
                          float* __restrict__ counts, int* __restrict__ voxid, int n) {
  const long stride = (long)gridDim.x * blockDim.x;
  for (long i = (long)blockIdx.x * blockDim.x + threadIdx.x; i < n; i += stride) {
    if (i + stride < n) __builtin_prefetch(&cluster[i + stride], 0, 1);  // global_prefetch
    int cl = cluster[i];
    int iv = rank[cl] - 1;        // ascending-rank inverse == jnp.unique inverse
    voxid[i] = iv;
    float4 p = pts[i];
    atomicAdd(&sums[(size_t)iv * 4 + 0], p.x);
    atomicAdd(&sums[(size_t)iv * 4 + 1], p.y);
    atomicAdd(&sums[(size_t)iv * 4 + 2], p.z);
    atomicAdd(&sums[(size_t)iv * 4 + 3], p.w);
    atomicAdd(&counts[iv], 1.0f);
  }
}

// ================= K7: voxel means, bcoords (int bits), bcenters =================
__global__ void k_finalize(float* __restrict__ out, const float* __restrict__ counts, int n) {
  int i = blockIdx.x * blockDim.x + threadIdx.x;
  if (i >= n) return;
  float4 s = ((float4*)out)[i];
  float cnt = counts[i];
  float r = 1.0f / fmaxf(cnt, 1.0f);
  float4 samp = make_float4(s.x * r, s.y * r, s.z * r, s.w * r);
  ((float4*)out)[i] = samp;
  float cx = truncf((samp.y - PLOX) / G1);
  float cy = truncf((samp.z - PLOY) / G1);
  float cz = truncf((samp.w - PLOZ) / G1);
  int4 bc = make_int4((int)rintf(samp.x), (int)rintf(cx), (int)rintf(cy), (int)rintf(cz));
  ((int4*)(out + 4 * (size_t)n))[i] = bc;
  float4 ctr = make_float4(samp.x,
                           cx * G1 + PLOX + 0.5f * G1,
                           cy * G1 + PLOY + 0.5f * G1,
                           cz * G1 + PLOZ + 0.5f * G1);
  ((float4*)(out + 8 * (size_t)n))[i] = ctr;
}

// ================= K8: per-point weights =================
__global__ void k_weight(const int* __restrict__ voxid, const float* __restrict__ counts,
                         float* __restrict__ weight, int n) {
  int i = blockIdx.x * blockDim.x + threadIdx.x;
  if (i >= n) return;
  weight[i] = 1.0f / counts[voxid[i]];
}

// ================= host launcher =================
extern "C" void kernel_launch(void* const* d_in, const int* in_sizes, int n_in,
                              void* d_out, int out_size, void* d_ws, size_t ws_size,
                              hipStream_t stream) {
  const float* pts = (const float*)d_in[0];
  const int N = in_sizes[0] / 4;  // 4,000,000

  // d_out layout (floats): sampled[4N] | bcoords[4N as int bits] | bcenter[4N]
  //                        | voxel_id[N as int bits] | weight[N]
  float* out = (float*)d_out;
  float* sums = out;
  int* voxid = (int*)(out + 12 * (size_t)N);
  float* weight = out + 13 * (size_t)N;

  // workspace: cluster[N] | counts[N] | flags/rank[M_PAD] | csum | coff | mm  (~80MB)
  uint8_t* ws = (uint8_t*)d_ws;
  int* cluster = (int*)ws;
  float* counts = (float*)(ws + (size_t)N * 4);
  int* flags = (int*)(ws + (size_t)N * 8);
  int* csum = (int*)(ws + (size_t)N * 8 + (size_t)M_PAD * 4);
  int* coff = csum + 2048;
  unsigned* mm = (unsigned*)(coff + 2048);

  k_init<<<2048, 256, 0, stream>>>(flags, counts, sums, mm, N);
  k_minmax<<<1024, 256, 0, stream>>>(pts, mm, N);
  k_cluster<<<4096, 256, 0, stream>>>((const float4*)pts, mm, cluster, flags, N);
  k_chunksum<<<NCHUNKS, 256, 0, stream>>>(flags, csum);
  k_scanchunks<<<1, 256, 0, stream>>>(csum, coff);
  k_rank<<<NCHUNKS, 256, 0, stream>>>(flags, coff);
  k_scatter<<<4096, 256, 0, stream>>>((const float4*)pts, cluster, flags, sums, counts, voxid, N);
  k_finalize<<<(N + 255) / 256, 256, 0, stream>>>(out, counts, N);
  k_weight<<<(N + 255) / 256, 256, 0, stream>>>(voxid, counts, weight, N);
}